// Stages1to5_68186900791928
// MI455X (gfx1250) — compile-verified
//
#include <hip/hip_runtime.h>
#include <math.h>

#define B_ 16
#define N_ 64
#define D_ 512
#define P_ 64
#define Q_ 16
#define CELL_ 1024

typedef __attribute__((ext_vector_type(2))) float v2f;
typedef __attribute__((ext_vector_type(4))) float v4f;
typedef __attribute__((ext_vector_type(8))) float v8f;

// ---------------------------------------------------------------------------
// Kernel 1: m and m_p, fully fused streaming kernel (memory-bound path).
// m[b,n,n2,pq]  = c[b,n,p] * d_ctx[b,n2,q]
// m_p           = s^3 * m + (s^2+s+1) * bias_m[n,n2,pq]
// Non-temporal stores: m/m_p are write-once, never re-read -> bypass L2
// so the 16MB bias_m stays resident for its 16x (per-batch) reuse.
// ---------------------------------------------------------------------------
__global__ void k_m_mp(const float* __restrict__ c, const float* __restrict__ d_ctx,
                       const float* __restrict__ bias_m, const float* __restrict__ alpha,
                       float* __restrict__ m_out, float* __restrict__ mp_out) {
  const float s  = 1.0f + alpha[0];
  const float s3 = s * s * s;
  const float sb = s * s + s + 1.0f;
  const int blk = blockIdx.x;          // b*4096 + n*64 + n2
  const int b   = blk >> 12;
  const int rem = blk & 4095;
  const int n   = rem >> 6;
  const int n2  = rem & 63;
  const int pq  = threadIdx.x << 2;    // 4 floats per thread (1024 / 256)
  const int p   = pq >> 4;
  const int q   = pq & 15;

  const float cv = c[(b * N_ + n) * P_ + p];
  const v4f d4   = *reinterpret_cast<const v4f*>(&d_ctx[(b * N_ + n2) * Q_ + q]);
  const v4f bias4 = *reinterpret_cast<const v4f*>(
      &bias_m[(size_t)(n * N_ + n2) * CELL_ + pq]);

  const v4f m4  = cv * d4;
  const v4f mp4 = s3 * m4 + sb * bias4;

  const size_t off = (size_t)blk * CELL_ + pq;
  __builtin_nontemporal_store(m4,  reinterpret_cast<v4f*>(&m_out[off]));
  __builtin_nontemporal_store(mp4, reinterpret_cast<v4f*>(&mp_out[off]));
}

// ---------------------------------------------------------------------------
// Kernel 2: dmean[b,q] = mean over n of d_ctx[b,n,q]   (exactly 256 outputs)
// ---------------------------------------------------------------------------
__global__ void k_dmean(const float* __restrict__ d_ctx, float* __restrict__ dmean) {
  const int t = threadIdx.x;           // b*16 + q
  const int b = t >> 4;
  const int q = t & 15;
  float s = 0.0f;
  for (int n = 0; n < N_; ++n) s += d_ctx[(b * N_ + n) * Q_ + q];
  dmean[t] = s * (1.0f / N_);
}

// ---------------------------------------------------------------------------
// Kernel 3: bias_mean[n,pq] = mean over n2 of bias_m[n,n2,pq]
// ---------------------------------------------------------------------------
__global__ void k_bias_mean(const float* __restrict__ bias_m,
                            float* __restrict__ bias_mean) {
  const int n  = blockIdx.x;
  const int pq = threadIdx.x << 2;
  v4f acc = {0.f, 0.f, 0.f, 0.f};
  for (int n2 = 0; n2 < N_; ++n2)
    acc += *reinterpret_cast<const v4f*>(&bias_m[(size_t)(n * N_ + n2) * CELL_ + pq]);
  *reinterpret_cast<v4f*>(&bias_mean[n * CELL_ + pq]) = acc * (1.0f / N_);
}

// ---------------------------------------------------------------------------
// Kernel 4: m_mean[b,n,pq] = s3 * c[b,n,p]*dmean[b,q] + sb * bias_mean[n,pq]
// (closed-form of m_p.mean(axis=2); avoids re-reading 268MB of m_p)
// ---------------------------------------------------------------------------
__global__ void k_mmean(const float* __restrict__ c, const float* __restrict__ dmean,
                        const float* __restrict__ bias_mean,
                        const float* __restrict__ alpha, float* __restrict__ mmean) {
  const float s  = 1.0f + alpha[0];
  const float s3 = s * s * s;
  const float sb = s * s + s + 1.0f;
  const int blk = blockIdx.x;          // b*64 + n
  const int b   = blk >> 6;
  const int n   = blk & 63;
  const int pq  = threadIdx.x << 2;
  const int p   = pq >> 4;
  const int q   = pq & 15;
  const float cv = c[(b * N_ + n) * P_ + p];
  const v4f dm = *reinterpret_cast<const v4f*>(&dmean[b * Q_ + q]);
  const v4f bm = *reinterpret_cast<const v4f*>(&bias_mean[n * CELL_ + pq]);
  const v4f r  = (s3 * cv) * dm + sb * bm;
  *reinterpret_cast<v4f*>(&mmean[(size_t)blk * CELL_ + pq]) = r;
}

// ---------------------------------------------------------------------------
// Kernel 5: q_proj/k_proj via V_WMMA_F32_16X16X4_F32 (full fp32 matrix path).
// q_proj[b,d,n] = sum_c m_mean[b,n,c] * w_q[d,c]   -> per-b GEMM 512x64 (K=1024)
// One wave per 16x16 output tile; B-fragment (m_mean) shared by both GEMMs.
// A 16x4 f32 layout: lanes 0-15 hold K=k,k+1; lanes 16-31 hold K=k+2,k+3.
// ---------------------------------------------------------------------------
__global__ void k_qk_wmma(const float* __restrict__ mmean,
                          const float* __restrict__ wq, const float* __restrict__ wk,
                          float* __restrict__ qout, float* __restrict__ kout) {
  const int lane = threadIdx.x;        // 0..31, EXEC all ones (WMMA requirement)
  const int half = lane >> 4;
  const int l    = lane & 15;
  const int tile_m = blockIdx.x * 16;  // over D
  const int tile_n = blockIdx.y * 16;  // over N
  const int b      = blockIdx.z;

  const float* __restrict__ mrow = &mmean[((size_t)b * N_ + (tile_n + l)) * CELL_];
  const float* __restrict__ qrow = &wq[(size_t)(tile_m + l) * CELL_];
  const float* __restrict__ krow = &wk[(size_t)(tile_m + l) * CELL_];

  v8f accq = {};
  v8f acck = {};
#pragma unroll 4
  for (int k = 0; k < CELL_; k += 4) {
    v2f a_q = { qrow[k + 2 * half], qrow[k + 2 * half + 1] };
    v2f a_k = { krow[k + 2 * half], krow[k + 2 * half + 1] };
    v2f bf  = { mrow[k + half],     mrow[k + 2 + half] };
    accq = __builtin_amdgcn_wmma_f32_16x16x4_f32(false, a_q, false, bf,
                                                 (short)0, accq, false, false);
    acck = __builtin_amdgcn_wmma_f32_16x16x4_f32(false, a_k, false, bf,
                                                 (short)0, acck, false, false);
  }

  // C/D 16x16 f32 layout: VGPR r, lanes 0-15 -> M=r, lanes 16-31 -> M=r+8
  const size_t base = (size_t)b * D_ * N_;
  const int nc = tile_n + l;
#pragma unroll
  for (int r = 0; r < 8; ++r) {
    const int mr = tile_m + r + 8 * half;
    qout[base + (size_t)mr * N_ + nc] = accq[r];
    kout[base + (size_t)mr * N_ + nc] = acck[r];
  }
}

// ---------------------------------------------------------------------------
// Kernel 6: cos_ab[b] and abn[b,:]  (one block per batch, LDS tree reduce)
// ---------------------------------------------------------------------------
__global__ void k_cos_abn(const float* __restrict__ a, const float* __restrict__ b,
                          float* __restrict__ cos_out, float* __restrict__ abn_out) {
  __shared__ float saa[256], sbb[256], sab[256];
  const int bb = blockIdx.x;
  const int t  = threadIdx.x;
  float aa = 0.f, bv2 = 0.f, ab = 0.f;
  for (int d = t; d < D_; d += 256) {
    const float av = a[bb * D_ + d];
    const float bvv = b[bb * D_ + d];
    aa += av * av; bv2 += bvv * bvv; ab += av * bvv;
  }
  saa[t] = aa; sbb[t] = bv2; sab[t] = ab;
  __syncthreads();
  for (int s = 128; s > 0; s >>= 1) {
    if (t < s) { saa[t] += saa[t + s]; sbb[t] += sbb[t + s]; sab[t] += sab[t + s]; }
    __syncthreads();
  }
  const float na = fmaxf(sqrtf(saa[0]), 1e-8f);
  const float nb = fmaxf(sqrtf(sbb[0]), 1e-8f);
  const float cosv = sab[0] / (na * nb);
  if (t == 0) cos_out[bb] = cosv;
  for (int d = t; d < D_; d += 256) {
    abn_out[bb * D_ + d] =
        tanhf(a[bb * D_ + d] * (1.0f - cosv)) + tanhf(b[bb * D_ + d] * cosv);
  }
}

// ---------------------------------------------------------------------------
// Kernel 7: fused tail per (b,n): c_tilde softmax, c_star, d_proj+abdj,
// logits -> p_hat -> c_dbl.  All tiny; one 256-thread block per (n,b).
// ---------------------------------------------------------------------------
__global__ void k_tail(const float* __restrict__ c, const float* __restrict__ d_ctx,
                       const float* __restrict__ abn,
                       const float* __restrict__ abd_w, const float* __restrict__ abd_b,
                       const float* __restrict__ ctx_w, const float* __restrict__ ctx_b,
                       const float* __restrict__ w1, const float* __restrict__ b1,
                       const float* __restrict__ w_prob, const float* __restrict__ b_prob,
                       float* __restrict__ abdj_out, float* __restrict__ ctil_out,
                       float* __restrict__ cstar_out, float* __restrict__ cdbl_out,
                       float* __restrict__ phat_out) {
  __shared__ float sc[P_];     // c_tilde row
  __shared__ float cs[P_];     // c_star row
  __shared__ float dcrow[Q_];  // d_ctx row
  __shared__ float red0[256], red1[256];
  __shared__ float ph[2];

  const int n  = blockIdx.x;
  const int bb = blockIdx.y;
  const int t  = threadIdx.x;

  if (t < Q_) dcrow[t] = d_ctx[(bb * N_ + n) * Q_ + t];
  if (t < P_) sc[t] = 64.0f * c[(bb * N_ + n) * P_ + t];
  __syncthreads();

  if (t == 0) {                       // softmax max over 64 (tiny, serial is fine)
    float mx = sc[0];
    for (int i = 1; i < P_; ++i) mx = fmaxf(mx, sc[i]);
    red0[0] = mx;
  }
  __syncthreads();
  const float mx = red0[0];
  if (t < P_) sc[t] = __expf(sc[t] - mx);
  __syncthreads();
  if (t == 0) {
    float s = 0.f;
    for (int i = 0; i < P_; ++i) s += sc[i];
    red0[0] = 1.0f / s;
  }
  __syncthreads();
  const float inv = red0[0];
  if (t < P_) {
    sc[t] *= inv;
    ctil_out[(bb * N_ + n) * P_ + t] = sc[t];
  }
  __syncthreads();

  if (t < P_) {                       // c_star
    float acc = b1[n * P_ + t];
    const float* wr = &w1[((size_t)n * P_ + t) * P_];
    for (int p = 0; p < P_; ++p) acc += sc[p] * wr[p];
    acc = fmaxf(acc, 0.0f);
    cs[t] = acc;
    cstar_out[(bb * N_ + n) * P_ + t] = acc;
  }
  __syncthreads();

  // abdj over D plus per-thread logit partials
  float l0 = 0.f, l1 = 0.f;
  for (int d = t; d < D_; d += 256) {
    float dp = ctx_b[d];
    const float* cw = &ctx_w[(size_t)d * Q_];
#pragma unroll
    for (int q = 0; q < Q_; ++q) dp += dcrow[q] * cw[q];
    const float v = tanhf(abd_w[n * D_ + d] * abn[bb * D_ + d] * dp + abd_b[n * D_ + d]);
    abdj_out[((size_t)bb * N_ + n) * D_ + d] = v;
    l0 += v * w_prob[(size_t)n * 1152 + 64 + d];
    l1 += v * w_prob[(size_t)n * 1152 + 576 + 64 + d];
  }
  if (t < P_) {                       // c_star part of `combined`
    l0 += cs[t] * w_prob[(size_t)n * 1152 + t];
    l1 += cs[t] * w_prob[(size_t)n * 1152 + 576 + t];
  }
  red0[t] = l0; red1[t] = l1;
  __syncthreads();
  for (int s = 128; s > 0; s >>= 1) {
    if (t < s) { red0[t] += red0[t + s]; red1[t] += red1[t + s]; }
    __syncthreads();
  }
  if (t == 0) {
    const float L0 = red0[0] + b_prob[n * 2 + 0];
    const float L1 = red1[0] + b_prob[n * 2 + 1];
    const float m = fmaxf(L0, L1);
    const float e0 = __expf(L0 - m), e1 = __expf(L1 - m);
    const float is = 1.0f / (e0 + e1);
    ph[0] = e0 * is; ph[1] = e1 * is;
    phat_out[(bb * N_ + n) * 2 + 0] = ph[0];
    phat_out[(bb * N_ + n) * 2 + 1] = ph[1];
  }
  __syncthreads();
  if (t < P_) {
    const float v = cs[t];
    cdbl_out[(bb * N_ + n) * P_ + t] = tanhf(ph[0] * v) + tanhf(ph[1] * v);
  }
}

// ---------------------------------------------------------------------------
extern "C" void kernel_launch(void* const* d_in, const int* in_sizes, int n_in,
                              void* d_out, int out_size, void* d_ws, size_t ws_size,
                              hipStream_t stream) {
  const float* a      = (const float*)d_in[0];
  const float* b      = (const float*)d_in[1];
  const float* c      = (const float*)d_in[2];
  const float* d_ctx  = (const float*)d_in[3];
  const float* alpha  = (const float*)d_in[4];
  const float* bias_m = (const float*)d_in[5];
  const float* w_q    = (const float*)d_in[6];
  const float* w_k    = (const float*)d_in[7];
  const float* abd_w  = (const float*)d_in[8];
  const float* abd_b  = (const float*)d_in[9];
  const float* ctx_w  = (const float*)d_in[10];
  const float* ctx_b  = (const float*)d_in[11];
  const float* w1     = (const float*)d_in[12];
  const float* b1     = (const float*)d_in[13];
  const float* w_prob = (const float*)d_in[14];
  const float* b_prob = (const float*)d_in[15];

  float* out = (float*)d_out;
  size_t off = 0;
  float* m_out    = out + off; off += (size_t)B_ * N_ * N_ * CELL_;  // m
  float* mp_out   = out + off; off += (size_t)B_ * N_ * N_ * CELL_;  // m_p
  float* q_out    = out + off; off += (size_t)B_ * D_ * N_;          // q_proj
  float* k_out    = out + off; off += (size_t)B_ * D_ * N_;          // k_proj
  float* cos_out  = out + off; off += B_;                            // cos_ab
  float* abn_out  = out + off; off += (size_t)B_ * D_;               // abn
  float* abdj_out = out + off; off += (size_t)B_ * N_ * D_;          // abdj
  float* ctil_out = out + off; off += (size_t)B_ * N_ * P_;          // c_tilde
  float* cstar_out= out + off; off += (size_t)B_ * N_ * P_;          // c_star
  float* cdbl_out = out + off; off += (size_t)B_ * N_ * P_;          // c_dbl
  float* phat_out = out + off; off += (size_t)B_ * N_ * 2;           // p_hat

  float* ws = (float*)d_ws;
  float* bias_mean = ws;                    // N*CELL        = 65536 floats
  float* dmean     = ws + 65536;            // B*Q           = 256 floats
  float* mmean     = ws + 65536 + 256;      // B*N*CELL      = 1048576 floats (16B aligned)

  k_dmean<<<1, 256, 0, stream>>>(d_ctx, dmean);
  k_bias_mean<<<N_, 256, 0, stream>>>(bias_m, bias_mean);
  k_m_mp<<<B_ * N_ * N_, 256, 0, stream>>>(c, d_ctx, bias_m, alpha, m_out, mp_out);
  k_mmean<<<B_ * N_, 256, 0, stream>>>(c, dmean, bias_mean, alpha, mmean);
  k_qk_wmma<<<dim3(D_ / 16, N_ / 16, B_), 32, 0, stream>>>(mmean, w_q, w_k, q_out, k_out);
  k_cos_abn<<<B_, 256, 0, stream>>>(a, b, cos_out, abn_out);
  k_tail<<<dim3(N_, B_), 256, 0, stream>>>(c, d_ctx, abn_out, abd_w, abd_b,
                                           ctx_w, ctx_b, w1, b1, w_prob, b_prob,
                                           abdj_out, ctil_out, cstar_out, cdbl_out,
                                           phat_out);
}